// Contour_to_distance_map_34763465294600
// MI455X (gfx1250) — compile-verified
//
#include <hip/hip_runtime.h>
#include <math.h>

typedef __attribute__((ext_vector_type(2))) float v2f;
typedef __attribute__((ext_vector_type(8))) float v8f;

#define MSIZE   256
#define NPTS    64
#define KSIGN   100000.0f
#define FEPS    1e-5f
#define FEPS2   1e-10f                 // FEPS^2
#define INV2PI  0.15915494309189535f   // 1/(2*pi)
#define TILES_PER_WAVE 8

#if __has_builtin(__builtin_amdgcn_tanhf)
#define FAST_TANH(x) __builtin_amdgcn_tanhf(x)
#else
#define FAST_TANH(x) tanhf(x)
#endif

#if __has_builtin(__builtin_amdgcn_rsqf)
#define FAST_RSQRT(x) __builtin_amdgcn_rsqf(x)
#else
#define FAST_RSQRT(x) rsqrtf(x)
#endif

__device__ __forceinline__ v2f pick2(float c0, float c1, float c2, float c3, int half) {
    v2f lo = {c0, c1};
    v2f hi = {c2, c3};
    return half ? hi : lo;
}

// One wave owns one contour instance and TILES_PER_WAVE tiles of 16 pixels.
// Per tile: 4 point-groups x 4 quantities = 16 x v_wmma_f32_16x16x4_f32.
__global__ void __launch_bounds__(256)
contour_dmap_kernel(const float* __restrict__ contour, // (4, 64, 2)
                    float* __restrict__ out,           // (4, 256, 256)
                    unsigned int* __restrict__ gmax)   // global max (float bits)
{
    const int lane = threadIdx.x & 31;
    const int wave = threadIdx.x >> 5;
    const int bn   = blockIdx.x;                 // contour instance 0..3
    const int base = blockIdx.y * 8 + wave;      // base tile 0..511
    const int col  = lane & 15;                  // matrix column / A row
    const int half = lane >> 4;                  // K-half select

    const float* __restrict__ C = contour + bn * (NPTS * 2);

    // ---------------- A fragments (per lane, contour-only; reused for all tiles) ----
    // A is 16x4 f32: lane holds row M = lane%16, K = vgpr + 2*(lane/16).
    v2f aN[4], aX[4], aD[4], aR[4];
#pragma unroll
    for (int t = 0; t < 4; ++t) {
        const int k  = 16 * t + col;
        const int kn = (k + 1) & (NPTS - 1);
        const float cx = C[2 * k],  cy = C[2 * k + 1];
        const float nx = C[2 * kn], ny = C[2 * kn + 1];
        // nd^2  coefficients vs f = [x, y, x^2+y^2, 1]
        aN[t] = pick2(-2.f * cx, -2.f * cy, 1.f, cx * cx + cy * cy, half);
        // cross coefficients
        aX[t] = pick2(ny - cy, cx - nx, 0.f, cy * nx - cx * ny, half);
        // dot   coefficients
        aD[t] = pick2(-(cx + nx), -(cy + ny), 1.f, cx * nx + cy * ny, half);
        // nr^2  coefficients (= nd^2 of next point)
        aR[t] = pick2(-2.f * nx, -2.f * ny, 1.f, nx * nx + ny * ny, half);
    }

    const v8f czero = {0.f, 0.f, 0.f, 0.f, 0.f, 0.f, 0.f, 0.f};

    float wave_max = 0.0f;

#pragma unroll 1
    for (int it = 0; it < TILES_PER_WAVE; ++it) {
        const int tile = base + 512 * it;        // 0..4095
        // ---------------- B fragment (per pixel feature vector) ----------------
        // B is 4x16 f32: lane holds column N = lane%16, K = vgpr + 2*(lane/16).
        const int p = tile * 16 + col;           // flat pixel index i*256+j
        const float x  = (float)(p >> 8) * (1.0f / MSIZE);
        const float y  = (float)(p & (MSIZE - 1)) * (1.0f / MSIZE);
        const float r2 = x * x + y * y;
        const v2f bf = pick2(x, y, r2, 1.0f, half);

        float acc   = 0.0f;
        float mind2 = 3.402823466e38f;

#pragma unroll
        for (int t = 0; t < 4; ++t) {
            // D = A x B : D[v], this lane -> point (16t + v + 8*half), pixel p
            v8f dN = __builtin_amdgcn_wmma_f32_16x16x4_f32(
                false, aN[t], false, bf, (short)0, czero, false, false);
            v8f dX = __builtin_amdgcn_wmma_f32_16x16x4_f32(
                false, aX[t], false, bf, (short)0, czero, false, false);
            v8f dD = __builtin_amdgcn_wmma_f32_16x16x4_f32(
                false, aD[t], false, bf, (short)0, czero, false, false);
            v8f dR = __builtin_amdgcn_wmma_f32_16x16x4_f32(
                false, aR[t], false, bf, (short)0, czero, false, false);

#pragma unroll
            for (int v = 0; v < 8; ++v) {
                const float nd2 = fmaxf(dN[v], 0.0f);
                const float cr  = dX[v];
                const float dt  = dD[v];
                const float nr2 = fmaxf(dR[v], 0.0f);

                mind2 = fminf(mind2, nd2);

                // max(sqrt(nd2),eps)*max(sqrt(nr2),eps) == sqrt(max(nd2,eps^2)*max(nr2,eps^2))
                const float den2 = fmaxf(nd2, FEPS2) * fmaxf(nr2, FEPS2);
                float ca = dt * FAST_RSQRT(den2);
                ca = fminf(fmaxf(ca, -1.0f + FEPS), 1.0f - FEPS);

                acc += FAST_TANH(KSIGN * cr) * acosf(ca);
            }
        }

        // Combine the two lane halves (each covered 32 of the 64 points).
        acc   = acc + __shfl_xor(acc, 16);
        mind2 = fminf(mind2, __shfl_xor(mind2, 16));

        const float winding = fabsf(acc) * INV2PI;
        const float prod    = winding * sqrtf(mind2);   // >= 0

        if (half == 0)
            out[bn * (MSIZE * MSIZE) + p] = prod;

        wave_max = fmaxf(wave_max, prod);
    }

    // Wave-level max, then one atomic per wave (uint-bit order == float order for >=0).
#pragma unroll
    for (int off = 1; off < 32; off <<= 1)
        wave_max = fmaxf(wave_max, __shfl_xor(wave_max, off));
    if (lane == 0)
        atomicMax(gmax, __float_as_uint(wave_max));
}

__global__ void init_max_kernel(unsigned int* gmax) {
    if (threadIdx.x == 0 && blockIdx.x == 0) gmax[0] = 0u;
}

__global__ void __launch_bounds__(256)
normalize_kernel(float* __restrict__ out, const unsigned int* __restrict__ gmax, int n) {
    const int i = blockIdx.x * blockDim.x + threadIdx.x;
    const float mv = __uint_as_float(gmax[0]);
    const float inv = (mv > 0.0f) ? (1.0f / mv) : 1.0f;
    if (i < n) out[i] *= inv;
}

extern "C" void kernel_launch(void* const* d_in, const int* in_sizes, int n_in,
                              void* d_out, int out_size, void* d_ws, size_t ws_size,
                              hipStream_t stream) {
    const float* contour = (const float*)d_in[0];   // (2,2,64,2) fp32
    float* out = (float*)d_out;                     // (2,2,256,256) fp32
    unsigned int* gmax = (unsigned int*)d_ws;

    init_max_kernel<<<1, 32, 0, stream>>>(gmax);

    // 4 contour instances x 4096 tiles of 16 pixels; 8 waves per block,
    // each wave loops over 8 tiles reusing its A fragments.
    dim3 grid(4, 64, 1);
    contour_dmap_kernel<<<grid, 256, 0, stream>>>(contour, out, gmax);

    const int n = 4 * MSIZE * MSIZE;
    normalize_kernel<<<(n + 255) / 256, 256, 0, stream>>>(out, gmax, n);
}